// Loss_88639535055378
// MI455X (gfx1250) — compile-verified
//
#include <hip/hip_runtime.h>

// ---------------------------------------------------------------------------
// YOLO box loss, single fused pass + tiny finalize.
//   loss = (5/512) * sum_a covered[a] * S[a]
//   S[a] = sum_{b,hw} gt_conf^2 * ||pred_box - gt_box||^2
//   covered[j] set if any (b,hw,a) argmax-IoU picks gt box j.
// Bandwidth-bound (~87 MB @ 23.3 TB/s => a few microseconds); FP32 scalar
// math, WMMA f32 16x16x4 used for the wave32 tree reduction.
// ---------------------------------------------------------------------------

#define A_COUNT 5
#define CH 125
#define HW 169
#define BATCH 512
#define NCELLS (BATCH * HW)          // 86528
#define THREADS 256
#define BLOCKS (NCELLS / THREADS)    // 338 (exact: 338*256 == 86528)
#define WAVES_PER_BLOCK (THREADS / 32)

typedef __attribute__((ext_vector_type(2))) float v2f;
typedef __attribute__((ext_vector_type(8))) float v8f;

// Full wave32 sum. WMMA path: A[m][*] = (v_m, 0, v_{m+16}, 0), B = ones(4x16)
// => D[m][n] = v_m + v_{m+16} for every n. D VGPR i holds row i (lanes 0-15)
// and row 8+i (lanes 16-31), so sum(c[0..7]) covers half the rows per lane;
// one xor-16 shuffle merges the halves. Deterministic.
__device__ __forceinline__ float wave32_sum(float v) {
#if __has_builtin(__builtin_amdgcn_wmma_f32_16x16x4_f32)
  v2f a; a.x = v;    a.y = 0.0f;
  v2f b; b.x = 1.0f; b.y = 1.0f;
  v8f c = {};
  c = __builtin_amdgcn_wmma_f32_16x16x4_f32(
      /*neg_a=*/false, a, /*neg_b=*/false, b,
      /*c_mod=*/(short)0, c, /*reuse_a=*/false, /*reuse_b=*/false);
  float s = c[0] + c[1] + c[2] + c[3] + c[4] + c[5] + c[6] + c[7];
  s += __shfl_xor(s, 16, 32);
  return s;
#else
  #pragma unroll
  for (int off = 16; off >= 1; off >>= 1) v += __shfl_xor(v, off, 32);
  return v;
#endif
}

__global__ __launch_bounds__(THREADS) void yolo_box_loss_kernel(
    const float* __restrict__ pred,     // [512,125,13,13]
    const float* __restrict__ tgt,      // [512,13,13,125]
    const float* __restrict__ anchors,  // [5,2]
    float* __restrict__ partial)        // [BLOCKS][8]: S[0..4], covbits, pad
{
  const int tid  = threadIdx.x;
  const int cell = blockIdx.x * THREADS + tid;          // always < NCELLS
  const int b    = cell / HW;
  const int hw   = cell - b * HW;

  const float* __restrict__ trow = tgt  + (size_t)b * (HW * CH) + (size_t)hw * CH;
  const float* __restrict__ prow = pred + (size_t)b * (CH * HW) + hw;

  float px[A_COUNT], py[A_COUNT], pw[A_COUNT], ph[A_COUNT];
  float gx[A_COUNT], gy[A_COUNT], gw[A_COUNT], gh[A_COUNT], gc[A_COUNT];
  float bx1[A_COUNT], by1[A_COUNT], bx2[A_COUNT], by2[A_COUNT], areaB[A_COUNT];

  #pragma unroll
  for (int a = 0; a < A_COUNT; ++a) {
    // prediction: channel plane stride HW, perfectly coalesced over hw
    const float* pc = prow + (size_t)(a * 25 + 21) * HW;
    float ox = pc[0];
    float oy = pc[HW];
    float ow = pc[2 * HW];
    float oh = pc[3 * HW];
    px[a] = 1.0f / (1.0f + expf(-ox));
    py[a] = 1.0f / (1.0f + expf(-oy));
    pw[a] = expf(ow);
    ph[a] = expf(oh);

    gc[a] = trow[a * 25 + 20];
    gx[a] = trow[a * 25 + 21];
    gy[a] = trow[a * 25 + 22];
    gw[a] = trow[a * 25 + 23];
    gh[a] = trow[a * 25 + 24];

    bx1[a] = gx[a] - gw[a] * 0.5f;
    by1[a] = gy[a] - gh[a] * 0.5f;
    bx2[a] = gx[a] + gw[a] * 0.5f;
    by2[a] = gy[a] + gh[a] * 0.5f;
    areaB[a] = (bx2[a] - bx1[a]) * (by2[a] - by1[a]);
  }

  float acc[A_COUNT];
  unsigned cov = 0u;

  #pragma unroll
  for (int a = 0; a < A_COUNT; ++a) {
    const float aw = anchors[2 * a];       // uniform -> scalar loads
    const float ah = anchors[2 * a + 1];
    const float ax1 = px[a] - aw * 0.5f;
    const float ay1 = py[a] - ah * 0.5f;
    const float ax2 = px[a] + aw * 0.5f;
    const float ay2 = py[a] + ah * 0.5f;
    const float areaA = (ax2 - ax1) * (ay2 - ay1);

    float best = 0.0f;
    int bestj = 0;
    #pragma unroll
    for (int j = 0; j < A_COUNT; ++j) {
      float iw = fminf(ax2, bx2[j]) - fmaxf(ax1, bx1[j]);
      float ih = fminf(ay2, by2[j]) - fmaxf(ay1, by1[j]);
      iw = fmaxf(iw, 0.0f);
      ih = fmaxf(ih, 0.0f);
      float inter = iw * ih;
      float iou = inter / (areaA + areaB[j] - inter + 1e-10f);
      if (j == 0 || iou > best) { best = iou; bestj = j; }  // first-max, as argmax
    }
    cov |= (1u << bestj);

    const float m = gc[a];
    const float dx = px[a] * m - gx[a] * m;
    const float dy = py[a] * m - gy[a] * m;
    const float dw = pw[a] * m - gw[a] * m;
    const float dh = ph[a] * m - gh[a] * m;
    acc[a] = dx * dx + dy * dy + dw * dw + dh * dh;
  }

  // ---- wave reduction (WMMA path; EXEC all-ones: no divergence above) ----
  float wsum[A_COUNT];
  #pragma unroll
  for (int a = 0; a < A_COUNT; ++a) wsum[a] = wave32_sum(acc[a]);
  #pragma unroll
  for (int off = 16; off >= 1; off >>= 1) cov |= __shfl_xor(cov, off, 32);

  // ---- deterministic block combine via LDS (no atomics) ----
  __shared__ float    lds_s[WAVES_PER_BLOCK][A_COUNT];
  __shared__ unsigned lds_c[WAVES_PER_BLOCK];
  const int wave = tid >> 5;
  const int lane = tid & 31;
  if (lane == 0) {
    #pragma unroll
    for (int a = 0; a < A_COUNT; ++a) lds_s[wave][a] = wsum[a];
    lds_c[wave] = cov;
  }
  __syncthreads();
  if (tid == 0) {
    float bsum[A_COUNT] = {0.f, 0.f, 0.f, 0.f, 0.f};
    unsigned bcov = 0u;
    for (int w = 0; w < WAVES_PER_BLOCK; ++w) {
      #pragma unroll
      for (int a = 0; a < A_COUNT; ++a) bsum[a] += lds_s[w][a];
      bcov |= lds_c[w];
    }
    float* dst = partial + (size_t)blockIdx.x * 8;
    #pragma unroll
    for (int a = 0; a < A_COUNT; ++a) dst[a] = bsum[a];
    dst[5] = __uint_as_float(bcov);
  }
}

__global__ void yolo_box_loss_finalize(const float* __restrict__ partial,
                                       float* __restrict__ out)
{
  if (threadIdx.x == 0 && blockIdx.x == 0) {
    float s[A_COUNT] = {0.f, 0.f, 0.f, 0.f, 0.f};
    unsigned cov = 0u;
    for (int blk = 0; blk < BLOCKS; ++blk) {       // fixed order: deterministic
      const float* p = partial + (size_t)blk * 8;
      #pragma unroll
      for (int a = 0; a < A_COUNT; ++a) s[a] += p[a];
      cov |= __float_as_uint(p[5]);
    }
    float total = 0.0f;
    #pragma unroll
    for (int a = 0; a < A_COUNT; ++a)
      if ((cov >> a) & 1u) total += s[a];
    out[0] = (5.0f / 512.0f) * total;              // LAMBDA_COORD / BATCH_SIZE
  }
}

extern "C" void kernel_launch(void* const* d_in, const int* in_sizes, int n_in,
                              void* d_out, int out_size, void* d_ws, size_t ws_size,
                              hipStream_t stream) {
  const float* pred = (const float*)d_in[0];  // [512,125,13,13] f32
  const float* tgt  = (const float*)d_in[1];  // [512,13,13,125] f32
  const float* anc  = (const float*)d_in[2];  // [5,2] f32
  float* out        = (float*)d_out;          // scalar f32
  float* partial    = (float*)d_ws;           // BLOCKS*8 floats = 10,816 B

  yolo_box_loss_kernel<<<BLOCKS, THREADS, 0, stream>>>(pred, tgt, anc, partial);
  yolo_box_loss_finalize<<<1, 32, 0, stream>>>(partial, out);
}